// MelCepstrumPostfiltering_55009941127443
// MI455X (gfx1250) — compile-verified
//
#include <hip/hip_runtime.h>

typedef __attribute__((ext_vector_type(2))) float v2f;
typedef __attribute__((ext_vector_type(8))) float v8f;

#define CEP    25      // cep_order+1
#define KPAD   28      // K padded to 7 WMMA steps of 4
#define NK     7
#define IRL    128
#define NSPEC  65      // rfft bins of length-128 FFT
#define NCOLS  130     // [s1(65) | s2(65)]
#define NT     9       // N tiles of 16 (144 cols, 130 used)
#define FPB    128     // frames per block (8 waves x 16)
#define DPI    3.14159265358979323846

// Constant B matrix, stored pre-swizzled in WMMA B-fragment order:
// g_pf[2*((t*NK+kk)*32 + lane) + j] = P[kk*4 + 2*(lane>>4) + j][t*16 + (lane&15)]
__device__ float g_pf[NT * NK * 64];

// ---------------------------------------------------------------------------
// Setup: build P = [M | M*diag(w)] (28 x 144, zero padded) in fragment order.
// M[k][m] = sum_n cos(2*pi*k*n/128) * A_FT[n][m],  A_FT = freqt(-alpha) matrix.
// ---------------------------------------------------------------------------
__global__ __launch_bounds__(256) void build_pf_kernel() {
    __shared__ double Ad[IRL][CEP];    // freqt matrix (128 x 25)
    __shared__ double Md[NSPEC][CEP];  // cosine-combined (65 x 25)
    const int tid = threadIdx.x;

    if (tid == 0) {
        const double a = -0.42;  // freqt uses -ALPHA
        double p = 1.0;
        for (int j = 0; j < CEP; ++j) { Ad[0][j] = p; p *= a; }
        Ad[1][0] = 0.0;
        p = 1.0;
        for (int j = 1; j < CEP; ++j) { Ad[1][j] = p * (double)j * (1.0 - a * a); p *= a; }
        for (int i = 2; i < IRL; ++i) {
            Ad[i][0] = 0.0;
            for (int j = 1; j < CEP; ++j)
                Ad[i][j] = Ad[i - 1][j - 1] + a * (Ad[i][j - 1] - Ad[i - 1][j]);
        }
    }
    __syncthreads();

    for (int idx = tid; idx < NSPEC * CEP; idx += 256) {
        const int k = idx / CEP, m = idx % CEP;
        const double w = 2.0 * DPI * (double)k / (double)IRL;
        double s = 0.0;
        for (int n = 0; n < IRL; ++n) s += cos(w * (double)n) * Ad[n][m];
        Md[k][m] = s;
    }
    __syncthreads();

    for (int i = tid; i < NT * NK * 32; i += 256) {
        const int lane = i & 31;
        const int kk   = (i >> 5) % NK;
        const int t    = (i >> 5) / NK;
        const int h    = lane >> 4;
        const int c    = t * 16 + (lane & 15);
        #pragma unroll
        for (int j = 0; j < 2; ++j) {
            const int row = kk * 4 + 2 * h + j;
            double v = 0.0;
            if (row < CEP) {
                const double wrow = (row < 2) ? 1.0 : 1.2;  // postfilter WEIGHT
                if (c < NSPEC)      v = Md[c][row];
                else if (c < NCOLS) v = Md[c - NSPEC][row] * wrow;
            }
            g_pf[2 * i + j] = (float)v;
        }
    }
}

// ---------------------------------------------------------------------------
// Main: streaming pass.  out = WEIGHT*mc ; out[...,0] += 0.5*log(e1/e2)
// e = sum_k bw_k * exp(2 * (M @ mc)_k)   via V_WMMA_F32_16X16X4_F32.
// ---------------------------------------------------------------------------
__global__ __launch_bounds__(256) void postfilter_kernel(
    const float* __restrict__ mc, float* __restrict__ out, int nframes) {
    __shared__ float  smc[FPB][KPAD];
    __shared__ float  sdelta[FPB];
    const int tid  = threadIdx.x;
    const int lane = tid & 31;
    const int wv   = tid >> 5;
    const long long base = (long long)blockIdx.x * FPB;

    // coalesced tile load, zero-padded K columns / tail frames
    for (int i = tid; i < FPB * KPAD; i += 256) {
        const int f = i / KPAD, m = i - f * KPAD;
        const long long gf = base + f;
        float v = 0.0f;
        if (m < CEP && gf < (long long)nframes) v = mc[gf * CEP + m];
        smc[f][m] = v;
    }
    __syncthreads();

    // ---- WMMA: Y(16 frames x 144 cols) = A(16x28) * P(28x144), K steps of 4
    const int fb   = wv * 16;
    const int mrow = lane & 15;
    const int h    = lane >> 4;

    v2f afrag[NK];  // A: VGPR0 = K=k0 | k0+2, VGPR1 = K=k0+1 | k0+3 (even LDS addr -> b64)
    #pragma unroll
    for (int kk = 0; kk < NK; ++kk)
        afrag[kk] = *(const v2f*)&smc[fb + mrow][kk * 4 + 2 * h];

    float acc1[8], acc2[8];
    #pragma unroll
    for (int r = 0; r < 8; ++r) { acc1[r] = 0.0f; acc2[r] = 0.0f; }

    for (int t = 0; t < NT; ++t) {
        v8f cacc = {};
        #pragma unroll
        for (int kk = 0; kk < NK; ++kk) {
            const v2f b = *(const v2f*)(g_pf + 2 * ((t * NK + kk) * 32 + lane));
            cacc = __builtin_amdgcn_wmma_f32_16x16x4_f32(
                false, afrag[kk], false, b, (short)0, cacc, false, false);
        }
        // bin weights for r[0]: DC/Nyquist once, interior bins twice, /128
        const int col = t * 16 + mrow;
        const float w1 = (col < NSPEC)
            ? ((col == 0 || col == NSPEC - 1) ? (1.0f / IRL) : (2.0f / IRL)) : 0.0f;
        const int c2 = col - NSPEC;
        const float w2 = (c2 >= 0 && c2 < NSPEC)
            ? ((c2 == 0 || c2 == NSPEC - 1) ? (1.0f / IRL) : (2.0f / IRL)) : 0.0f;

        union { v8f v; float f[8]; } u;
        u.v = cacc;
        #pragma unroll
        for (int r = 0; r < 8; ++r) {
            const float e = __expf(2.0f * u.f[r]);
            acc1[r] += w1 * e;
            acc2[r] += w2 * e;
        }
    }

    // reduce across the 16-lane half (frame r + 8*h lives in that half's VGPR r)
    #pragma unroll
    for (int r = 0; r < 8; ++r) {
        float s1 = acc1[r], s2 = acc2[r];
        #pragma unroll
        for (int m = 1; m <= 8; m <<= 1) {
            s1 += __shfl_xor(s1, m, 32);
            s2 += __shfl_xor(s2, m, 32);
        }
        if (mrow == 0)
            sdelta[fb + h * 8 + r] = 0.5f * (__logf(s1) - __logf(s2));
    }
    __syncthreads();

    // coalesced output: out = WEIGHT*mc (+ delta on m==0)
    for (int i = tid; i < FPB * CEP; i += 256) {
        const int f = i / CEP, m = i - f * CEP;
        const long long gf = base + f;
        if (gf < (long long)nframes) {
            float v = smc[f][m] * ((m < 2) ? 1.0f : 1.2f);
            if (m == 0) v += sdelta[f];
            out[gf * CEP + m] = v;
        }
    }
}

extern "C" void kernel_launch(void* const* d_in, const int* in_sizes, int n_in,
                              void* d_out, int out_size, void* d_ws, size_t ws_size,
                              hipStream_t stream) {
    (void)n_in; (void)d_ws; (void)ws_size; (void)out_size;
    const float* mc = (const float*)d_in[0];
    float* out = (float*)d_out;
    const int nframes = in_sizes[0] / CEP;

    build_pf_kernel<<<1, 256, 0, stream>>>();
    const int nblocks = (nframes + FPB - 1) / FPB;
    if (nblocks > 0)
        postfilter_kernel<<<nblocks, 256, 0, stream>>>(mc, out, nframes);
}